// Discretizer_67886253081389
// MI455X (gfx1250) — compile-verified
//
#include <hip/hip_runtime.h>

// ---------------------------------------------------------------------------
// CDNA5 / gfx1250 implementation of the Discretizer reference.
// All heavy math (6x conv64->64 GEMMs + VQ distance GEMM) runs on
// V_WMMA_F32_16X16X4_F32 (fp32 WMMA -> matches reference numerics).
// ---------------------------------------------------------------------------

typedef __attribute__((ext_vector_type(2))) float v2f;
typedef __attribute__((ext_vector_type(8))) float v8f;

#define LSEQ   8192
#define NBATCH 32
#define NPOS   (NBATCH * LSEQ)   // 262144 positions
#define DIMC   64
#define NCODE  512
#define KCONV  192               // 3 taps * 64 channels

__device__ __forceinline__ int clampi(int v, int lo, int hi) {
    return v < lo ? lo : (v > hi ? hi : v);
}

// ---------------------------------------------------------------------------
// Prep: rearrange six (64,64,3) OIH weight tensors into [K=192][N=64] GEMM
// layout; build codebook^T [64][512] and per-code squared norms.
// ---------------------------------------------------------------------------
__global__ void prep_kernel(const float* __restrict__ w0, const float* __restrict__ w1,
                            const float* __restrict__ w2, const float* __restrict__ w3,
                            const float* __restrict__ w4, const float* __restrict__ w5,
                            const float* __restrict__ cb,
                            float* __restrict__ W2,      // [6][192*64]
                            float* __restrict__ cbT,     // [64][512]
                            float* __restrict__ cnorm)   // [512]
{
    int id = blockIdx.x * blockDim.x + threadIdx.x;
    const int NW = 6 * KCONV * DIMC;           // 73728
    const int NT = NCODE * DIMC;               // 32768
    if (id < NW) {
        const float* ws[6] = {w0, w1, w2, w3, w4, w5};
        int li = id % (KCONV * DIMC);
        int l  = id / (KCONV * DIMC);
        int oc = li & 63;
        int c  = li >> 6;          // K index: tap*64 + ic
        int ic = c & 63;
        int k  = c >> 6;
        W2[l * (KCONV * DIMC) + li] = ws[l][oc * 192 + ic * 3 + k];  // w[oc][ic][k]
    } else if (id < NW + NT) {
        int j = id - NW;           // flat [d][n]
        int d = j >> 9, n = j & 511;
        cbT[j] = cb[n * 64 + d];
    } else if (id < NW + NT + NCODE) {
        int n = id - (NW + NT);
        float s = 0.f;
        for (int d = 0; d < 64; ++d) { float v = cb[n * 64 + d]; s += v * v; }
        cnorm[n] = s;
    }
}

// ---------------------------------------------------------------------------
// Upsample (x4, half-pixel linear, edge clamp) fused with conv1 (1 -> 64).
// One thread per output position.
// ---------------------------------------------------------------------------
__global__ void up_conv1_kernel(const float* __restrict__ x,   // [32][2048]
                                const float* __restrict__ w,   // [64][1][3]
                                const float* __restrict__ bias,// [64]
                                float* __restrict__ out)       // [NPOS][64]
{
    int pos = blockIdx.x * blockDim.x + threadIdx.x;
    if (pos >= NPOS) return;
    int b = pos >> 13, t = pos & (LSEQ - 1);
    const float* xb = x + b * 2048;
    float u[3];
#pragma unroll
    for (int j = 0; j < 3; ++j) {
        int tt = clampi(t + j - 1, 0, LSEQ - 1);
        float src = tt * 0.25f - 0.375f;       // (p+0.5)/4 - 0.5
        float fi = floorf(src);
        int i0 = (int)fi;
        float fw = src - fi;
        int ia = clampi(i0, 0, 2047);
        int ib = clampi(i0 + 1, 0, 2047);
        u[j] = xb[ia] * (1.0f - fw) + xb[ib] * fw;
    }
    float* op = out + (size_t)pos * 64;
#pragma unroll 8
    for (int oc = 0; oc < 64; ++oc)
        op[oc] = w[oc * 3 + 0] * u[0] + w[oc * 3 + 1] * u[1] + w[oc * 3 + 2] * u[2] + bias[oc];
}

// ---------------------------------------------------------------------------
// Conv1d 64->64 k=3 replicate-pad as im2col GEMM on V_WMMA_F32_16X16X4_F32.
// Block = 256 threads = 8 waves; wave -> one 16x16 (M-pos x N-chan) tile.
// waves: ntile = w&3 (4 x 16 channels), mtile = blk*2 + (w>>2).
// K loop: 48 steps of K=4. A fragment per ISA layout: lane M=lane%16,
// K0 = 2*(lane>=16); B symmetric; C: lane,r -> M=r+8*(lane>=16), N=lane%16.
// ---------------------------------------------------------------------------
__global__ void conv_gemm64_kernel(const float* __restrict__ in,   // [NPOS][64]
                                   const float* __restrict__ Wk,   // [192][64]
                                   const float* __restrict__ bias, // [64]
                                   float* __restrict__ out)        // [NPOS][64]
{
    int wave  = threadIdx.x >> 5;
    int lane  = threadIdx.x & 31;
    int half  = lane >> 4;                 // 0 or 1
    int lm    = lane & 15;
    int ntile = wave & 3;
    int m0    = (blockIdx.x * 2 + (wave >> 2)) * 16;   // tile never crosses batch (8192%16==0)
    int rowM  = m0 + lm;                   // A row this lane feeds
    int b     = rowM >> 13;
    int tb    = rowM & (LSEQ - 1);
    int n     = ntile * 16 + lm;           // B/C column

    v8f acc = {};
#pragma unroll 4
    for (int kk = 0; kk < 48; ++kk) {
        int c0  = kk * 4 + half * 2;       // K column (and c0+1); never crosses a tap
        int tap = c0 >> 6;
        int ic  = c0 & 63;
        int t   = clampi(tb + tap - 1, 0, LSEQ - 1);
        const float* ap = in + ((size_t)(b * LSEQ + t)) * 64 + ic;
        v2f a = { ap[0], ap[1] };
        v2f bf = { Wk[c0 * 64 + n], Wk[(c0 + 1) * 64 + n] };
        acc = __builtin_amdgcn_wmma_f32_16x16x4_f32(false, a, false, bf,
                                                    (short)0, acc, false, false);
    }
    float bs = bias[n];
#pragma unroll
    for (int r = 0; r < 8; ++r) {
        int M = m0 + r + half * 8;
        out[(size_t)M * 64 + n] = acc[r] + bs;
    }
}

// ---------------------------------------------------------------------------
// Vector quantize: per wave one 16-position tile. G = z . c via WMMA over 32
// codebook N-tiles; argmin of (cnorm - 2G) per row (first-index ties);
// writes zq rows; deterministic per-block loss partial.
// ---------------------------------------------------------------------------
__global__ void vq_kernel(const float* __restrict__ z,      // [NPOS][64]
                          const float* __restrict__ cbT,    // [64][512]
                          const float* __restrict__ cb,     // [512][64]
                          const float* __restrict__ cnorm,  // [512]
                          float* __restrict__ zq,           // [NPOS][64]
                          float* __restrict__ lossPart)     // [gridDim.x]
{
    __shared__ float sloss[16];
    int wave = threadIdx.x >> 5;
    int lane = threadIdx.x & 31;
    int half = lane >> 4;
    int lm   = lane & 15;
    int m0   = (blockIdx.x * 8 + wave) * 16;

    // Cache this lane's A-row slice (its 32 of the row's 64 floats).
    float arow[32];
    const float* zr = z + (size_t)(m0 + lm) * 64;
#pragma unroll
    for (int kk = 0; kk < 16; ++kk) {
        arow[2 * kk]     = zr[kk * 4 + half * 2];
        arow[2 * kk + 1] = zr[kk * 4 + half * 2 + 1];
    }
    // |z|^2 for row lm (combine the two half-lanes).
    float zn = 0.f;
#pragma unroll
    for (int i = 0; i < 32; ++i) zn += arow[i] * arow[i];
    zn += __shfl_xor(zn, 16, 32);          // lanes lm and lm+16 now hold row lm's norm

    float minv[8]; int mini[8];
#pragma unroll
    for (int r = 0; r < 8; ++r) { minv[r] = 3.4e38f; mini[r] = 0; }

    for (int nt = 0; nt < 32; ++nt) {
        int n = nt * 16 + lm;
        v8f acc = {};
#pragma unroll
        for (int kk = 0; kk < 16; ++kk) {
            v2f a = { arow[2 * kk], arow[2 * kk + 1] };
            int K0 = kk * 4 + half * 2;
            v2f bf = { cbT[K0 * 512 + n], cbT[(K0 + 1) * 512 + n] };
            acc = __builtin_amdgcn_wmma_f32_16x16x4_f32(false, a, false, bf,
                                                        (short)0, acc, false, false);
        }
        float cn = cnorm[n];
#pragma unroll
        for (int r = 0; r < 8; ++r) {
            float val = cn - 2.0f * acc[r];
            if (val < minv[r]) { minv[r] = val; mini[r] = n; }  // n ascending -> first min kept
        }
    }
    // Reduce argmin across the 16 lanes of each half (N dimension), lowest index on ties.
#pragma unroll
    for (int r = 0; r < 8; ++r) {
        float v = minv[r]; int i = mini[r];
        for (int s = 1; s < 16; s <<= 1) {
            float ov = __shfl_xor(v, s, 32);
            int   oi = __shfl_xor(i, s, 32);
            if (ov < v || (ov == v && oi < i)) { v = ov; i = oi; }
        }
        minv[r] = v; mini[r] = i;
    }
    // Write zq rows + accumulate loss = sum_r (minv[r] + |z_row|^2).
    float s = 0.f;
#pragma unroll
    for (int r = 0; r < 8; ++r) {
        int M = m0 + r + half * 8;
        int idx = mini[r];
        const float4* crow = (const float4*)(cb + (size_t)idx * 64 + lm * 4);
        *(float4*)(zq + (size_t)M * 64 + lm * 4) = *crow;
        float znr = __shfl(zn, r + half * 8, 32);   // row M's |z|^2 lives at lane M%16
        s += minv[r] + znr;
    }
    if (lm == 0) sloss[wave * 2 + half] = s;
    __syncthreads();
    if (threadIdx.x == 0) {
        float t = 0.f;
#pragma unroll
        for (int i = 0; i < 16; ++i) t += sloss[i];   // fixed order -> deterministic
        lossPart[blockIdx.x] = t;
    }
}

// ---------------------------------------------------------------------------
// Final conv 64->1 (replicate pad); trailing resize to same length is identity.
// Thread 0 also folds the deterministic loss partials into d_out[NPOS].
// ---------------------------------------------------------------------------
__global__ void out4_final_kernel(const float* __restrict__ act,  // [NPOS][64]
                                  const float* __restrict__ w,    // [1][64][3]
                                  const float* __restrict__ bias, // [1]
                                  const float* __restrict__ lossPart, int nPart,
                                  float* __restrict__ dout)       // [NPOS + 1]
{
    int pos = blockIdx.x * blockDim.x + threadIdx.x;
    if (pos < NPOS) {
        int b = pos >> 13, t = pos & (LSEQ - 1);
        float s = bias[0];
#pragma unroll
        for (int k = 0; k < 3; ++k) {
            int tt = clampi(t + k - 1, 0, LSEQ - 1);
            const float* ar = act + ((size_t)(b * LSEQ + tt)) * 64;
#pragma unroll 8
            for (int ic = 0; ic < 64; ++ic) s += ar[ic] * w[ic * 3 + k];
        }
        dout[pos] = s;
    }
    if (blockIdx.x == 0 && threadIdx.x == 0) {
        float acc = 0.f;
        for (int i = 0; i < nPart; ++i) acc += lossPart[i];
        dout[NPOS] = acc * (1.0f / ((float)NPOS * 64.0f));  // mean over b*l*d
    }
}

// ---------------------------------------------------------------------------
extern "C" void kernel_launch(void* const* d_in, const int* in_sizes, int n_in,
                              void* d_out, int out_size, void* d_ws, size_t ws_size,
                              hipStream_t stream) {
    const float* xhat_c = (const float*)d_in[0];
    const float* in_w1  = (const float*)d_in[4];  const float* in_b1 = (const float*)d_in[5];
    const float* in_w2  = (const float*)d_in[6];  const float* in_b2 = (const float*)d_in[7];
    const float* in_w3  = (const float*)d_in[8];  const float* in_b3 = (const float*)d_in[9];
    const float* in_w4  = (const float*)d_in[10]; const float* in_b4 = (const float*)d_in[11];
    const float* cb     = (const float*)d_in[12];
    const float* o_w1   = (const float*)d_in[13]; const float* o_b1 = (const float*)d_in[14];
    const float* o_w2   = (const float*)d_in[15]; const float* o_b2 = (const float*)d_in[16];
    const float* o_w3   = (const float*)d_in[17]; const float* o_b3 = (const float*)d_in[18];
    const float* o_w4   = (const float*)d_in[19]; const float* o_b4 = (const float*)d_in[20];
    float* dout = (float*)d_out;

    // Workspace layout (floats)
    float* ws = (float*)d_ws;
    float* actA  = ws;                                // 16,777,216
    float* actB  = actA + (size_t)NPOS * 64;          // 16,777,216
    float* W2    = actB + (size_t)NPOS * 64;          // 6 * 12288
    float* cbT   = W2 + 6 * KCONV * DIMC;             // 32768
    float* cnorm = cbT + NCODE * DIMC;                // 512
    float* lossP = cnorm + NCODE;                     // 2048

    const int prepThreads = 6 * KCONV * DIMC + NCODE * DIMC + NCODE;   // 107008
    prep_kernel<<<(prepThreads + 255) / 256, 256, 0, stream>>>(
        in_w2, in_w3, in_w4, o_w1, o_w2, o_w3, cb, W2, cbT, cnorm);

    up_conv1_kernel<<<NPOS / 256, 256, 0, stream>>>(xhat_c, in_w1, in_b1, actA);

    // in-convs 2..4
    conv_gemm64_kernel<<<NPOS / 32, 256, 0, stream>>>(actA, W2 + 0 * 12288, in_b2, actB);
    conv_gemm64_kernel<<<NPOS / 32, 256, 0, stream>>>(actB, W2 + 1 * 12288, in_b3, actA);
    conv_gemm64_kernel<<<NPOS / 32, 256, 0, stream>>>(actA, W2 + 2 * 12288, in_b4, actB);

    // vector quantize: z=actB -> zq=actA
    const int vqBlocks = NPOS / 128;                  // 2048
    vq_kernel<<<vqBlocks, 256, 0, stream>>>(actB, cbT, cb, cnorm, actA, lossP);

    // out-convs 1..3
    conv_gemm64_kernel<<<NPOS / 32, 256, 0, stream>>>(actA, W2 + 3 * 12288, o_b1, actB);
    conv_gemm64_kernel<<<NPOS / 32, 256, 0, stream>>>(actB, W2 + 4 * 12288, o_b2, actA);
    conv_gemm64_kernel<<<NPOS / 32, 256, 0, stream>>>(actA, W2 + 5 * 12288, o_b3, actB);

    // final 64->1 conv + identity resize + loss finalize
    out4_final_kernel<<<NPOS / 256, 256, 0, stream>>>(actB, o_w4, o_b4, lossP, vqBlocks, dout);
}